// RHN_audio_encoder_74646531604574
// MI455X (gfx1250) — compile-verified
//
#include <hip/hip_runtime.h>
#include <math.h>

// ---------------------------------------------------------------------------
// Types for CDNA5 WMMA f32 16x16x4:  A(16x4)=2 VGPR, B(4x16)=2 VGPR, C/D=8 VGPR
// ---------------------------------------------------------------------------
typedef float v2f __attribute__((ext_vector_type(2)));
typedef float v8f __attribute__((ext_vector_type(8)));

__device__ __forceinline__ v8f wmma_f32(v2f a, v2f b, v8f c) {
  // (neg_a, A, neg_b, B, c_mod, C, reuse_a, reuse_b)
  return __builtin_amdgcn_wmma_f32_16x16x4_f32(false, a, false, b, (short)0, c,
                                               false, false);
}

#define HID   1024
#define HID2  2048
#define BATCH 32
#define TLEN  1022            // (2048 - 6)/2 + 1
#define MTOT  (TLEN * BATCH)  // 32704 = 64 * 511

// ---------------------------------------------------------------------------
// Conv2d frontend: x (32,1,40,2048), w (64,1,40,6), stride (1,2) -> (t,b,c)
// out[(t*32+b)*64 + c]
// ---------------------------------------------------------------------------
__global__ __launch_bounds__(256) void conv_front(
    const float* __restrict__ x, const float* __restrict__ w,
    const float* __restrict__ cb, float* __restrict__ yout) {
  const int t   = blockIdx.x;                      // 0..1021
  const int idx = blockIdx.y * 256 + threadIdx.x;  // 0..2047
  const int b = idx >> 6, c = idx & 63;
  const float* xb = x + (size_t)b * 40 * 2048 + 2 * t;
  const float* wc = w + c * 240;
  float acc = cb[c];
  for (int f = 0; f < 40; ++f) {
    const float* xr = xb + f * 2048;
    const float* wr = wc + f * 6;
#pragma unroll
    for (int k = 0; k < 6; ++k) acc += xr[k] * wr[k];
  }
  yout[((size_t)t * BATCH + b) * 64 + c] = acc;
}

// ---------------------------------------------------------------------------
// Generic fp32 WMMA GEMM: C[M,N] = act(A[M,K] @ B[K,N] + bias[N])
// Block tile 64x128, 8 waves -> 32x32 each (2x2 WMMA frags).
// LDS tiles pair-interleaved over k ([k/2][col][k&1]) -> each WMMA fragment
// is one aligned ds_load_b64. All 16 fragments of a K-tile are loaded before
// the 16 WMMAs issue, so loads overlap the matrix pipe.
// Requires M%64==0, N%128==0, K%16==0.  act: 0=none, 1=tanh
// ---------------------------------------------------------------------------
__global__ __launch_bounds__(256) void gemm_wmma(
    const float* __restrict__ A, const float* __restrict__ B,
    const float* __restrict__ bias, float* __restrict__ C,
    int M, int N, int K, int act) {
  __shared__ __align__(16) float As2[8 * 64 * 2];    // [kp][m][par]  4 KB
  __shared__ __align__(16) float Bs2[8 * 128 * 2];   // [kp][n][par]  8 KB
  const int tid  = threadIdx.x;
  const int lane = tid & 31;
  const int l15  = lane & 15;
  const int hs   = lane >> 4;   // 0: k-slice {0,1}, 1: k-slice {2,3}
  const int w    = tid >> 5;    // wave 0..7
  const int wm   = (w & 1) * 32;
  const int wn   = (w >> 1) * 32;
  const int m0 = blockIdx.x * 64;
  const int n0 = blockIdx.y * 128;

  v8f acc[2][2];
  acc[0][0] = {}; acc[0][1] = {}; acc[1][0] = {}; acc[1][1] = {};

  // staging thread mappings (pointer-stepped, no per-iter 64-bit muls)
  const int ar  = tid >> 2;          // A: row 0..63
  const int ac  = (tid & 3) * 4;     // A: k 0,4,8,12
  const int bkp = tid >> 6;          // B: k-pair base 0..3
  const int bn  = (tid & 63) * 2;    // B: n 0..126 (even)
  const float* aptr  = A + (size_t)(m0 + ar) * K + ac;
  const float* bptr0 = B + (size_t)(2 * bkp) * N + n0 + bn;
  const float* bptr1 = bptr0 + N;
  const size_t bstep = (size_t)8 * N;  // 8 rows between the two it-chunks

  for (int k0 = 0; k0 < K; k0 += 16) {
    {  // stage A 64x16 -> pair-interleaved
      const float4 av = *(const float4*)aptr;
      *(float2*)&As2[((ac >> 1) + 0) * 128 + ar * 2] = make_float2(av.x, av.y);
      *(float2*)&As2[((ac >> 1) + 1) * 128 + ar * 2] = make_float2(av.z, av.w);
    }
    {  // stage B 16x128 -> pair-interleaved ([kp][n][par]), b128 LDS stores
#pragma unroll
      for (int it = 0; it < 2; ++it) {
        const float2 g0 = *(const float2*)(bptr0 + (size_t)it * bstep);
        const float2 g1 = *(const float2*)(bptr1 + (size_t)it * bstep);
        *(float4*)&Bs2[(bkp + it * 4) * 256 + bn * 2] =
            make_float4(g0.x, g1.x, g0.y, g1.y);
      }
    }
    if (k0 + 16 < K) {  // prefetch next tiles into cache
      __builtin_prefetch(aptr + 16, 0, 0);
      __builtin_prefetch(bptr0 + (size_t)16 * N, 0, 0);
    }
    aptr += 16;
    bptr0 += (size_t)16 * N;
    bptr1 += (size_t)16 * N;
    __syncthreads();

    v2f a0f[4], a1f[4], b0f[4], b1f[4];
#pragma unroll
    for (int i = 0; i < 4; ++i) {  // load ALL fragments of this K-tile first
      const int kp = i * 2 + hs;
      a0f[i] = *(const v2f*)&As2[kp * 128 + (wm + l15) * 2];
      a1f[i] = *(const v2f*)&As2[kp * 128 + (wm + 16 + l15) * 2];
      b0f[i] = *(const v2f*)&Bs2[kp * 256 + (wn + l15) * 2];
      b1f[i] = *(const v2f*)&Bs2[kp * 256 + (wn + 16 + l15) * 2];
    }
#pragma unroll
    for (int i = 0; i < 4; ++i) {
      acc[0][0] = wmma_f32(a0f[i], b0f[i], acc[0][0]);
      acc[0][1] = wmma_f32(a0f[i], b1f[i], acc[0][1]);
      acc[1][0] = wmma_f32(a1f[i], b0f[i], acc[1][0]);
      acc[1][1] = wmma_f32(a1f[i], b1f[i], acc[1][1]);
    }
    __syncthreads();
  }
#pragma unroll
  for (int mi = 0; mi < 2; ++mi)
#pragma unroll
    for (int ni = 0; ni < 2; ++ni) {
      const int n = n0 + wn + ni * 16 + l15;
      const float bv = bias[n];
#pragma unroll
      for (int r = 0; r < 8; ++r) {
        const int m = m0 + wm + mi * 16 + r + hs * 8;
        float v = acc[mi][ni][r] + bv;
        if (act == 1) v = tanhf(v);
        C[(size_t)m * N + n] = v;
      }
    }
}

// ---------------------------------------------------------------------------
// Pre-pack recurrent weights R (2,1024,2048) into pair-interleaved layout
// Rp[((l*512 + k/2)*2048 + n)*2 + (k&1)] so scan B-fragments are single
// coalesced global_load_b64.
// ---------------------------------------------------------------------------
__global__ __launch_bounds__(256) void pack_R(const float* __restrict__ R,
                                              float* __restrict__ Rp) {
  const int idx = blockIdx.x * 256 + threadIdx.x;  // 0 .. 2*512*2048-1
  const int n  = idx & 2047;
  const int kp = (idx >> 11) & 511;
  const int l  = idx >> 20;
  const float* src = R + ((size_t)l * 1024 + kp * 2) * 2048 + n;
  *(float2*)(Rp + (size_t)idx * 2) = make_float2(src[0], src[2048]);
}

// ---------------------------------------------------------------------------
// Device-wide barrier for the persistent scan kernel (generation counter).
// ctrl[0] = arrival count, ctrl[1] = generation.
// ---------------------------------------------------------------------------
__device__ __forceinline__ void grid_barrier(unsigned* ctrl, unsigned nb) {
  __syncthreads();
  if (threadIdx.x == 0) {
    __threadfence();  // release this block's global stores
    const unsigned g =
        __hip_atomic_load(&ctrl[1], __ATOMIC_RELAXED, __HIP_MEMORY_SCOPE_AGENT);
    const unsigned a = __hip_atomic_fetch_add(&ctrl[0], 1u, __ATOMIC_ACQ_REL,
                                              __HIP_MEMORY_SCOPE_AGENT);
    if (a == nb - 1u) {
      __hip_atomic_store(&ctrl[0], 0u, __ATOMIC_RELAXED,
                         __HIP_MEMORY_SCOPE_AGENT);
      __hip_atomic_fetch_add(&ctrl[1], 1u, __ATOMIC_RELEASE,
                             __HIP_MEMORY_SCOPE_AGENT);
    } else {
      while (__hip_atomic_load(&ctrl[1], __ATOMIC_ACQUIRE,
                               __HIP_MEMORY_SCOPE_AGENT) == g)
        __builtin_amdgcn_s_sleep(2);
    }
  }
  __syncthreads();
  __threadfence();  // make cross-block stores visible to every wave
}

// ---------------------------------------------------------------------------
// RHN recurrent scan for one layer (persistent, 16 blocks x 256 threads).
// 128 waves; wave u: m0 = (u>>6)*16, col-group g = u&63 computes h-pre tile
// [m0:m0+16, g*16:g*16+16] and t-pre tile [.., 1024+g*16 ..], fuses gates,
// writes new state. Two grid barriers per time step (one per micro-layer).
// State staged in LDS pair-interleaved: Ss2[(k/2)*32 + m'*2 + (k&1)].
// K-loop processes 16-K chunks: 12 fragment loads issued first, then 8 WMMAs
// on two split accumulator chains (halved accumulate dependency depth).
// ---------------------------------------------------------------------------
__global__ __launch_bounds__(256) void rhn_scan(
    const float* __restrict__ xp,   // (T*32, 2048) precomputed input proj
    const float* __restrict__ Rp,   // pair-interleaved (2,512,2048,2)
    const float* __restrict__ bR,   // (2, 2048)
    float* s0, float* s1,           // ping-pong state 32x1024 each
    float* __restrict__ yout,       // (T*32, 1024)
    unsigned* ctrl, int T) {
  __shared__ __align__(16) float Ss2[512 * 16 * 2];  // 64 KB
  const int tid  = threadIdx.x;
  const int lane = tid & 31;
  const int l15  = lane & 15;
  const int hs   = lane >> 4;
  const int u    = blockIdx.x * 8 + (tid >> 5);  // 0..127
  const int m0   = (u >> 6) * 16;                // 0 or 16 (uniform per block)
  const int g    = u & 63;
  const int colh = g * 16;
  const int colt = HID + g * 16;
  const unsigned NB = gridDim.x;

  // zero-init state
  for (int i = blockIdx.x * 256 + tid; i < BATCH * HID; i += (int)NB * 256) {
    s0[i] = 0.f;
    s1[i] = 0.f;
  }
  grid_barrier(ctrl, NB);

  float* scur = s0;
  float* snxt = s1;

  const int sr  = tid & 15;        // staging: row within 16-row slab
  const int scb = (tid >> 4) * 4;  // staging: k base

  for (int t = 0; t < T; ++t) {
    for (int l = 0; l < 2; ++l) {
      {  // stage scur rows [m0, m0+16) into LDS, pair-interleaved over k
        const float* srow = scur + (size_t)(m0 + sr) * HID;
#pragma unroll 4
        for (int j = 0; j < 16; ++j) {
          const int c = scb + j * 64;
          const float4 v = *(const float4*)(srow + c);
          *(float2*)&Ss2[((c >> 1) + 0) * 32 + sr * 2] = make_float2(v.x, v.y);
          *(float2*)&Ss2[((c >> 1) + 1) * 32 + sr * 2] = make_float2(v.z, v.w);
        }
      }
      __syncthreads();

      const float* Rlp = Rp + (size_t)l * 512 * HID2 * 2;
      // per-lane fragment bases; advance by constant stride per 16-K chunk
      const float* rh0 = Rlp + (size_t)(colh + l15) * 2 +
                         (size_t)hs * (HID2 * 2);
      const float* rt0 = Rlp + (size_t)(colt + l15) * 2 +
                         (size_t)hs * (HID2 * 2);
      v8f ah0 = {}, ah1 = {}, at0 = {}, at1 = {};
      for (int k0 = 0; k0 < HID; k0 += 16) {
        v2f af[4], bhf[4], btf[4];
#pragma unroll
        for (int i = 0; i < 4; ++i) {  // issue all loads of the chunk first
          const int kp = (k0 >> 1) + i * 2 + hs;
          af[i]  = *(const v2f*)&Ss2[kp * 32 + l15 * 2];
          bhf[i] = *(const v2f*)(rh0 + (size_t)i * (2 * HID2 * 2));
          btf[i] = *(const v2f*)(rt0 + (size_t)i * (2 * HID2 * 2));
        }
        rh0 += (size_t)8 * HID2 * 2;
        rt0 += (size_t)8 * HID2 * 2;
        // two independent accumulate chains (even/odd slices)
        ah0 = wmma_f32(af[0], bhf[0], ah0);
        at0 = wmma_f32(af[0], btf[0], at0);
        ah1 = wmma_f32(af[1], bhf[1], ah1);
        at1 = wmma_f32(af[1], btf[1], at1);
        ah0 = wmma_f32(af[2], bhf[2], ah0);
        at0 = wmma_f32(af[2], btf[2], at0);
        ah1 = wmma_f32(af[3], bhf[3], ah1);
        at1 = wmma_f32(af[3], btf[3], at1);
      }
      const v8f ah = ah0 + ah1;
      const v8f at = at0 + at1;

      // epilogue: bias (+xp for micro-layer 0), gates, state update
      const float biash = bR[l * HID2 + colh + l15];
      const float biast = bR[l * HID2 + colt + l15];
      const float* xpt = xp + (size_t)t * BATCH * HID2;
      const int jc = colh + l15;  // state column this lane owns
#pragma unroll
      for (int r = 0; r < 8; ++r) {
        const int m = m0 + r + hs * 8;
        float ph = ah[r] + biash;
        float pt = at[r] + biast;
        if (l == 0) {
          ph += xpt[(size_t)m * HID2 + colh + l15];
          pt += xpt[(size_t)m * HID2 + colt + l15];
        }
        const float hv   = tanhf(ph);
        const float tg   = 1.f / (1.f + __expf(-pt));
        const float sold = Ss2[(jc >> 1) * 32 + (m - m0) * 2 + (jc & 1)];
        const float snew = hv * tg + sold * (1.f - tg);
        snxt[(size_t)m * HID + jc] = snew;
        if (l == 1)
          yout[((size_t)t * BATCH + m) * HID + jc] = snew;
      }
      grid_barrier(ctrl, NB);
      float* tmp = scur; scur = snxt; snxt = tmp;
    }
  }
}

// ---------------------------------------------------------------------------
// Attention pooling: softmax over t per (b,h), weighted sum of y.
// logits/y layout: [(t*32+b)*1024 + h]; out[b*1024 + h].
// ---------------------------------------------------------------------------
__global__ __launch_bounds__(256) void att_pool(
    const float* __restrict__ logits, const float* __restrict__ y,
    float* __restrict__ out, int T) {
  const int idx = blockIdx.x * 256 + threadIdx.x;  // 0..32767
  const int b = idx >> 10, h = idx & 1023;
  const size_t stride = (size_t)BATCH * HID;
  const float* lp = logits + (size_t)b * HID + h;
  const float* yp = y + (size_t)b * HID + h;
  float mx = -INFINITY;
  for (int t = 0; t < T; ++t) mx = fmaxf(mx, lp[(size_t)t * stride]);
  float se = 0.f;
  for (int t = 0; t < T; ++t) se += __expf(lp[(size_t)t * stride] - mx);
  const float inv = 1.f / se;
  float acc = 0.f;
  for (int t = 0; t < T; ++t)
    acc += __expf(lp[(size_t)t * stride] - mx) * inv * yp[(size_t)t * stride];
  out[idx] = acc;
}

// ---------------------------------------------------------------------------
// Host launcher
// ---------------------------------------------------------------------------
extern "C" void kernel_launch(void* const* d_in, const int* in_sizes, int n_in,
                              void* d_out, int out_size, void* d_ws,
                              size_t ws_size, hipStream_t stream) {
  (void)in_sizes; (void)n_in; (void)out_size; (void)ws_size;
  const float* x     = (const float*)d_in[0];
  const float* convw = (const float*)d_in[1];
  const float* convb = (const float*)d_in[2];
  const float* W[4]  = {(const float*)d_in[3],  (const float*)d_in[7],
                        (const float*)d_in[11], (const float*)d_in[15]};
  const float* bW[4] = {(const float*)d_in[4],  (const float*)d_in[8],
                        (const float*)d_in[12], (const float*)d_in[16]};
  const float* R[4]  = {(const float*)d_in[5],  (const float*)d_in[9],
                        (const float*)d_in[13], (const float*)d_in[17]};
  const float* bR[4] = {(const float*)d_in[6],  (const float*)d_in[10],
                        (const float*)d_in[14], (const float*)d_in[18]};
  const float* att_wh = (const float*)d_in[19];
  const float* att_bh = (const float*)d_in[20];
  const float* att_wo = (const float*)d_in[21];
  const float* att_bo = (const float*)d_in[22];

  char* ws = (char*)d_ws;
  size_t off = 0;
  auto take = [&](size_t bytes) {
    void* p = ws + off;
    off = (off + bytes + 255) & ~(size_t)255;
    return p;
  };
  unsigned* ctrl  = (unsigned*)take(256);
  float* convout  = (float*)take((size_t)MTOT * 64 * 4);
  float* y0       = (float*)take((size_t)MTOT * HID * 4);
  float* y1       = (float*)take((size_t)MTOT * HID * 4);
  float* xp       = (float*)take((size_t)MTOT * HID2 * 4);  // reused as logits
  float* ebuf     = (float*)take((size_t)MTOT * 128 * 4);
  float* rpack    = (float*)take((size_t)2 * HID * HID2 * 4);  // 16 MB
  float* s0       = (float*)take((size_t)BATCH * HID * 4);
  float* s1       = (float*)take((size_t)BATCH * HID * 4);

  hipMemsetAsync(ctrl, 0, 256, stream);

  // conv frontend
  conv_front<<<dim3(TLEN, 8), 256, 0, stream>>>(x, convw, convb, convout);

  // 4 RHN layers: input projection GEMM + weight pre-pack + persistent scan
  const float* layer_in[4] = {convout, y0, y1, y0};
  float* layer_out[4]      = {y0, y1, y0, y1};
  const int layer_k[4]     = {64, HID, HID, HID};
  for (int L = 0; L < 4; ++L) {
    gemm_wmma<<<dim3(MTOT / 64, HID2 / 128), 256, 0, stream>>>(
        layer_in[L], W[L], bW[L], xp, MTOT, HID2, layer_k[L], 0);
    pack_R<<<dim3((2 * 512 * 2048) / 256), 256, 0, stream>>>(R[L], rpack);
    rhn_scan<<<16, 256, 0, stream>>>(xp, rpack, bR[L], s0, s1, layer_out[L],
                                     ctrl, TLEN);
  }

  // attention: e = tanh(y @ att_wh + bh), logits = e @ att_wo + bo
  gemm_wmma<<<dim3(MTOT / 64, 1), 256, 0, stream>>>(
      y1, att_wh, att_bh, ebuf, MTOT, 128, HID, 1);
  gemm_wmma<<<dim3(MTOT / 64, HID / 128), 256, 0, stream>>>(
      ebuf, att_wo, att_bo, xp, MTOT, HID, 128, 0);

  // softmax over time + weighted pooling -> (32, 1024)
  att_pool<<<dim3(BATCH * HID / 256), 256, 0, stream>>>(xp, y1, (float*)d_out,
                                                        TLEN);
}